// ArrayWiseSpatialLikelihoodGrid_20074677141779
// MI455X (gfx1250) — compile-verified
//
#include <hip/hip_runtime.h>
#include <math.h>

// Problem constants (match reference)
#define BB 16
#define KK 4
#define MM 4
#define NN 8192
#define PP 6
#define NG 128
#define THK 10
#define SRATE 16000.0f
#define CSND 343.0f

typedef float v2f __attribute__((ext_vector_type(2)));
typedef float v8f __attribute__((ext_vector_type(8)));

__device__ __constant__ int c_II[PP] = {0, 0, 0, 1, 1, 2};
__device__ __constant__ int c_JJ[PP] = {1, 2, 3, 2, 3, 3};

// ---------------------------------------------------------------------------
// Four-step complex FFT of length 8192 for one workgroup of 256 threads.
//   x[n], n = 512*n1 + n2  (linear in LDS xr/xi)
//   Step 1: Y[k1][n2] = sum_n1 W16^{sigma*k1*n1} x[n1][n2]      <- WMMA f32 16x16x4
//   Step 2: Y *= W8192^{sigma*k1*n2}                             (twiddle)
//   Step 3: 16 independent radix-2 512-point FFTs over n2        (in LDS)
//   Result: Z[k1][k2] at yr/yi[512*k1 + k2]; X[k1 + 16*k2] = Z[k1][k2]
// sigma = -1 forward, +1 inverse (caller scales by 1/N for inverse).
// ---------------------------------------------------------------------------
__device__ void fft8192(float* xr, float* xi, float* yr, float* yi,
                        int tid, float sigma) {
  const float TWO_PI = 6.2831853071795864769f;
  const int lane = tid & 31;
  const int wv   = tid >> 5;   // 8 waves per block
  const int lh   = lane >> 4;  // lane half (0/1)
  const int ln   = lane & 15;

  // ---- Step 1 + 2: radix-16 stage as 16x16x512 complex matmul via WMMA ----
  for (int t = wv; t < 32; t += 8) {       // 32 tiles of 16 columns, 4 per wave
    const int n2b = t << 4;
    v8f cr = {0.f, 0.f, 0.f, 0.f, 0.f, 0.f, 0.f, 0.f};
    v8f ci = {0.f, 0.f, 0.f, 0.f, 0.f, 0.f, 0.f, 0.f};
#pragma unroll
    for (int ch = 0; ch < 4; ++ch) {       // K = 16 in chunks of 4
      const int kb = ch << 2;
      v2f ar, ai, nai, br, bi;
#pragma unroll
      for (int j = 0; j < 2; ++j) {
        const int n1 = kb + j + (lh << 1);          // A/B K index per ISA layout
        const int ph = (ln * n1) & 15;              // DFT16 phase mod 16
        float s, c;
        __sincosf((TWO_PI / 16.0f) * (float)ph, &s, &c);
        ar[j]  = c;                                  // Re(DFT16[k1=ln][n1])
        ai[j]  = sigma * s;                          // Im(...), sign per direction
        nai[j] = -ai[j];                             // f32 WMMA has no A-neg
        const int src = (n1 << 9) + n2b + ln;        // x[512*n1 + n2]
        br[j] = xr[src];
        bi[j] = xi[src];
      }
      // Yr += Ar*Br - Ai*Bi ; Yi += Ar*Bi + Ai*Br
      cr = __builtin_amdgcn_wmma_f32_16x16x4_f32(false, ar,  false, br, (short)0, cr, false, false);
      cr = __builtin_amdgcn_wmma_f32_16x16x4_f32(false, nai, false, bi, (short)0, cr, false, false);
      ci = __builtin_amdgcn_wmma_f32_16x16x4_f32(false, ar,  false, bi, (short)0, ci, false, false);
      ci = __builtin_amdgcn_wmma_f32_16x16x4_f32(false, ai,  false, br, (short)0, ci, false, false);
    }
    // twiddle W8192^{sigma*k1*n2} and scatter into Y buffers
#pragma unroll
    for (int v = 0; v < 8; ++v) {
      const int k1 = v + (lh << 3);                  // C/D row per ISA layout
      const int n2 = n2b + ln;
      float s, c;
      __sincosf(sigma * (TWO_PI / 8192.0f) * (float)(k1 * n2), &s, &c);
      const float zr = cr[v] * c - ci[v] * s;
      const float zi = cr[v] * s + ci[v] * c;
      yr[(k1 << 9) + n2] = zr;
      yi[(k1 << 9) + n2] = zi;
    }
  }
  __syncthreads();

  // ---- Step 3: 16 rows x 512-point radix-2 DIT FFT in place ----
  // bit-reversal permutation (9 bits) within each row
  for (int i = tid; i < 8192; i += 256) {
    const int rb = i & ~511;
    const int n  = i & 511;
    const int r  = (int)(__brev((unsigned)n) >> 23);
    if (n < r) {
      const int a = rb + n, b = rb + r;
      float tr = yr[a]; yr[a] = yr[b]; yr[b] = tr;
      float ti = yi[a]; yi[a] = yi[b]; yi[b] = ti;
    }
  }
  __syncthreads();
  for (int h = 1; h < 512; h <<= 1) {
    const float astep = sigma * (TWO_PI * 0.5f) / (float)h;  // pi/h
    for (int j = tid; j < 4096; j += 256) {        // 16 rows * 256 butterflies
      const int rb  = (j >> 8) << 9;
      const int bf  = j & 255;
      const int pos = bf & (h - 1);
      const int i0  = rb + ((bf ^ pos) << 1) + pos;
      const int i1  = i0 + h;
      float s, c;
      __sincosf(astep * (float)pos, &s, &c);
      const float xr1 = yr[i1], xi1 = yi[i1];
      const float tr = c * xr1 - s * xi1;
      const float ti = c * xi1 + s * xr1;
      const float xr0 = yr[i0], xi0 = yi[i0];
      yr[i1] = xr0 - tr; yi[i1] = xi0 - ti;
      yr[i0] = xr0 + tr; yi[i0] = xi0 + ti;
    }
    __syncthreads();
  }
}

// ---------------------------------------------------------------------------
// Kernel 1: forward FFT of every (b,k,m) channel (real input, imag = 0).
// Signal staged into LDS with CDNA5 async-to-LDS B128 copies (ASYNCcnt).
// spec layout: interleaved complex, spec[2*(ch*8192 + k)] = Re, +1 = Im.
// ---------------------------------------------------------------------------
__global__ void k_fwd_fft(const float* __restrict__ signal, float* __restrict__ spec) {
  extern __shared__ float smem[];
  float* xr = smem;
  float* xi = smem + 8192;
  float* yr = smem + 16384;
  float* yi = smem + 24576;
  const int tid = threadIdx.x;
  const int ch  = blockIdx.x;                 // 0..255
  const long base = (long)ch * NN;

  // Async DMA: 8192 floats = 2048 x b128, 8 per thread. LDS byte address is
  // the low 32 bits of the generic shared-space pointer (aperture mapping).
  {
    const float* gsrc = signal + base;
    const unsigned lds_base = (unsigned)(unsigned long long)(const void*)xr;
#pragma unroll
    for (int it = 0; it < 8; ++it) {
      const int idx  = tid + it * 256;        // b128 chunk id
      const int voff = idx * 16;              // byte offset into signal row
      const unsigned ldsa = lds_base + (unsigned)voff;
      asm volatile("global_load_async_to_lds_b128 %0, %1, %2"
                   :: "v"(ldsa), "v"(voff), "s"(gsrc)
                   : "memory");
    }
    for (int i = tid; i < NN; i += 256) xi[i] = 0.0f;
    asm volatile("s_wait_asynccnt 0x0" ::: "memory");
  }
  __syncthreads();

  fft8192(xr, xi, yr, yi, tid, -1.0f);

  // write out coalesced in k: X[k] = Z[k1][k2], k = k1 + 16*k2
  for (int k = tid; k < NN; k += 256) {
    const int k1 = k & 15, k2 = k >> 4;
    const int li = (k1 << 9) + k2;
    const long o = 2l * (base + k);
    spec[o]     = yr[li];
    spec[o + 1] = yi[li];
  }
}

// ---------------------------------------------------------------------------
// Kernel 2: per mic pair: PHAT cross-spectrum + inverse FFT + fftshift -> cc
// ---------------------------------------------------------------------------
__global__ void k_cross_ifft(const float* __restrict__ spec, float* __restrict__ cc) {
  extern __shared__ float smem[];
  float* xr = smem;
  float* xi = smem + 8192;
  float* yr = smem + 16384;
  float* yi = smem + 24576;
  const int tid = threadIdx.x;
  const int pr  = blockIdx.x;                 // 0..383
  const int bk  = pr / PP;
  const int p   = pr - bk * PP;
  const long chi = (long)(bk * MM + c_II[p]) * NN;
  const long chj = (long)(bk * MM + c_JJ[p]) * NN;

  for (int n = tid; n < NN; n += 256) {
    const float a = spec[2l * (chi + n)],     b = spec[2l * (chi + n) + 1];
    const float c = spec[2l * (chj + n)],     d = spec[2l * (chj + n) + 1];
    // Xi * conj(Xj)
    const float re = a * c + b * d;
    const float im = b * c - a * d;
    const float den = fmaxf(sqrtf(re * re + im * im), 1e-9f);
    xr[n] = re / den;
    xi[n] = im / den;
  }
  __syncthreads();
  fft8192(xr, xi, yr, yi, tid, +1.0f);        // inverse (unscaled)
  const float inv_n = 1.0f / (float)NN;
  float* dst = cc + (long)pr * NN;
  // coalesced in k; fftshift puts lag 0 at N/2
  for (int k = tid; k < NN; k += 256) {
    const int k1 = k & 15, k2 = k >> 4;
    dst[(k + NN / 2) & (NN - 1)] = yr[(k1 << 9) + k2] * inv_n;
  }
}

// ---------------------------------------------------------------------------
// Kernel 3: SRP grid. One thread per grid point; 63M gathers, cc lives in L2.
// out (B,K,G*G) unnormalized; flat index = gx*G + gy.
// ---------------------------------------------------------------------------
__global__ void k_srp(const float* __restrict__ cc, const float* __restrict__ mic,
                      const float* __restrict__ room, float* __restrict__ out) {
  const int b = blockIdx.z;
  const int k = blockIdx.y;
  const int gid = blockIdx.x * 256 + threadIdx.x;   // 0..16383
  const int gx = gid >> 7;
  const int gy = gid & 127;
  const float px = room[b * 3 + 0] * ((float)gx * (1.0f / (NG - 1)));
  const float py = room[b * 3 + 1] * ((float)gy * (1.0f / (NG - 1)));

  const long ccb = (long)(b * KK + k) * PP * NN;
#pragma unroll
  for (int p = 0; p < PP; ++p)                       // warm the gather window
    __builtin_prefetch(cc + ccb + (long)p * NN + NN / 2, 0, 0);

  float d[MM];
#pragma unroll
  for (int m = 0; m < MM; ++m) {
    const long mo = ((long)(b * KK + k) * MM + m) * 3;
    const float dx = px - mic[mo + 0];
    const float dy = py - mic[mo + 1];
    d[m] = sqrtf(dx * dx + dy * dy);
  }

  float acc = 0.0f;
#pragma unroll
  for (int p = 0; p < PP; ++p) {
    const float tdoa = (d[c_II[p]] - d[c_JJ[p]]) * (1.0f / CSND);
    const int delay = (int)rintf(tdoa * SRATE) + NN / 2;   // round-half-even
    const float* row = cc + ccb + (long)p * NN;
#pragma unroll
    for (int o = 0; o < THK; ++o) {
      int idx = delay + o - THK / 2;
      idx = idx < 0 ? 0 : (idx > NN - 1 ? NN - 1 : idx);
      acc += row[idx];
    }
  }
  out[(long)(b * KK + k) * (NG * NG) + gid] = acc;
}

// ---------------------------------------------------------------------------
// Kernel 4: per-(b,k) max over the 16384-point grid
// ---------------------------------------------------------------------------
__global__ void k_maxred(const float* __restrict__ g, float* __restrict__ mx) {
  __shared__ float sm[256];
  const int bk = blockIdx.x;
  const float* p = g + (long)bk * (NG * NG);
  float m = -3.402823466e38f;
  for (int i = threadIdx.x; i < NG * NG; i += 256) m = fmaxf(m, p[i]);
  sm[threadIdx.x] = m;
  __syncthreads();
  for (int s = 128; s > 0; s >>= 1) {
    if (threadIdx.x < s) sm[threadIdx.x] = fmaxf(sm[threadIdx.x], sm[threadIdx.x + s]);
    __syncthreads();
  }
  if (threadIdx.x == 0) mx[bk] = sm[0];
}

// ---------------------------------------------------------------------------
// Kernel 5: normalize in place
// ---------------------------------------------------------------------------
__global__ void k_norm(float* __restrict__ out, const float* __restrict__ mx) {
  const long i = (long)blockIdx.x * 256 + threadIdx.x;   // 1,048,576 total
  out[i] = out[i] / mx[i >> 14];
}

// ---------------------------------------------------------------------------
extern "C" void kernel_launch(void* const* d_in, const int* in_sizes, int n_in,
                              void* d_out, int out_size, void* d_ws, size_t ws_size,
                              hipStream_t stream) {
  const float* signal = (const float*)d_in[0];   // (B,K,M,N)
  const float* mic    = (const float*)d_in[1];   // (B,K,M,3)
  const float* room   = (const float*)d_in[2];   // (B,3)
  float* out = (float*)d_out;                    // (B,K,G*G)
  float* ws  = (float*)d_ws;

  float* spec   = ws;                                     // 256*8192*2 floats
  float* cc     = spec + (long)BB * KK * MM * NN * 2;     // 384*8192 floats
  float* maxbuf = cc + (long)BB * KK * PP * NN;           // 64 floats

  const size_t lds = 4 * 8192 * sizeof(float);            // 128 KB dynamic LDS

  k_fwd_fft  <<<BB * KK * MM, 256, lds, stream>>>(signal, spec);
  k_cross_ifft<<<BB * KK * PP, 256, lds, stream>>>(spec, cc);
  k_srp      <<<dim3((NG * NG) / 256, KK, BB), 256, 0, stream>>>(cc, mic, room, out);
  k_maxred   <<<BB * KK, 256, 0, stream>>>(out, maxbuf);
  k_norm     <<<(BB * KK * NG * NG) / 256, 256, 0, stream>>>(out, maxbuf);
}